// MoEFSCIL_7473243095603
// MI455X (gfx1250) — compile-verified
//
#include <hip/hip_runtime.h>
#include <hip/hip_bf16.h>
#include <math.h>

#define B_    32
#define DIM   768
#define E_    4
#define K_    4
#define N_    16
#define R_    48
#define L_    49
#define BL    (B_ * L_)        // 1568
#define CPROJ (R_ + 2 * N_)    // 80
#define D2    (2 * DIM)        // 1536

typedef __attribute__((ext_vector_type(2))) float v2f;
typedef __attribute__((ext_vector_type(8))) float v8f;

// D = A(16x4, f32) * B(4x16, f32) + C(16x16, f32)  -- full-precision CDNA5 WMMA
__device__ __forceinline__ v8f wmma_f32x4(v2f a, v2f b, v8f c) {
  return __builtin_amdgcn_wmma_f32_16x16x4_f32(false, a, false, b, (short)0, c,
                                               false, false);
}

// xs index permutation per scan direction k (7x7 spatial)
__device__ __forceinline__ int scan_idx(int k, int l) {
  int t = (l % 7) * 7 + l / 7;
  if (k == 0) return l;
  if (k == 1) return 48 - l;
  if (k == 2) return t;
  return 48 - t;
}

__device__ __forceinline__ float silu_f(float x) {
  return x / (1.0f + __expf(-x));
}

// ---------------------------------------------------------------------------
// Gate stage 1: per-sample logits = mean_HW(x) @ gate_w + gate_b, softmax
// ---------------------------------------------------------------------------
__global__ __launch_bounds__(256) void gate1_kernel(
    const float* __restrict__ x, const float* __restrict__ gw,
    const float* __restrict__ gb, float* __restrict__ raw) {
  const int b = blockIdx.x;
  const int t = threadIdx.x;
  __shared__ float red[256][E_];
  float acc[E_] = {0.f, 0.f, 0.f, 0.f};
  for (int d = t; d < DIM; d += 256) {
    float xm = 0.f;
    for (int l = 0; l < L_; ++l) xm += x[(size_t)(b * L_ + l) * DIM + d];
    xm *= (1.0f / (float)L_);
#pragma unroll
    for (int e = 0; e < E_; ++e) acc[e] += xm * gw[d * E_ + e];
  }
#pragma unroll
  for (int e = 0; e < E_; ++e) red[t][e] = acc[e];
  __syncthreads();
  for (int off = 128; off > 0; off >>= 1) {
    if (t < off) {
#pragma unroll
      for (int e = 0; e < E_; ++e) red[t][e] += red[t + off][e];
    }
    __syncthreads();
  }
  if (t == 0) {
    float lg[E_], mx = -1e30f;
#pragma unroll
    for (int e = 0; e < E_; ++e) {
      lg[e] = red[0][e] + gb[e];
      mx = fmaxf(mx, lg[e]);
    }
    float s = 0.f;
#pragma unroll
    for (int e = 0; e < E_; ++e) { lg[e] = __expf(lg[e] - mx); s += lg[e]; }
#pragma unroll
    for (int e = 0; e < E_; ++e) raw[b * E_ + e] = lg[e] / s;
  }
}

// ---------------------------------------------------------------------------
// Gate stage 2: top-2 routing, capacity scaling, aux loss, final weights
// ---------------------------------------------------------------------------
__global__ __launch_bounds__(32) void gate2_kernel(
    const float* __restrict__ raw, float* __restrict__ wout,
    float* __restrict__ auxp) {
  const int b = threadIdx.x;  // 32 threads, one per sample
  __shared__ float masked_s[B_][E_];
  __shared__ float colsum[E_], rawmean[E_], maskmean[E_];
  float r[E_];
#pragma unroll
  for (int e = 0; e < E_; ++e) r[e] = raw[b * E_ + e];
  int i0 = 0;
#pragma unroll
  for (int e = 1; e < E_; ++e) if (r[e] > r[i0]) i0 = e;
  int i1 = -1;
#pragma unroll
  for (int e = 0; e < E_; ++e)
    if (e != i0 && (i1 < 0 || r[e] > r[i1])) i1 = e;
#pragma unroll
  for (int e = 0; e < E_; ++e)
    masked_s[b][e] = (e == i0 || e == i1) ? r[e] : 0.f;
  __syncthreads();
  if (b < E_) {
    float cs = 0.f, rm = 0.f, mm = 0.f;
    for (int bb = 0; bb < B_; ++bb) {
      cs += masked_s[bb][b];
      rm += raw[bb * E_ + b];
      mm += (masked_s[bb][b] > 0.f) ? 1.f : 0.f;
    }
    colsum[b] = cs;
    rawmean[b] = rm / (float)B_;
    maskmean[b] = mm / (float)B_;
  }
  __syncthreads();
  const float capacity = 40.0f;  // int(1.25 * 32)
  float gs[E_];
#pragma unroll
  for (int e = 0; e < E_; ++e)
    gs[e] = masked_s[b][e] / (colsum[e] + 1e-6f) * capacity;
  int j0 = 0;
#pragma unroll
  for (int e = 1; e < E_; ++e) if (gs[e] > gs[j0]) j0 = e;
  int j1 = -1;
#pragma unroll
  for (int e = 0; e < E_; ++e)
    if (e != j0 && (j1 < 0 || gs[e] > gs[j1])) j1 = e;
#pragma unroll
  for (int e = 0; e < E_; ++e)
    wout[b * E_ + e] = (e == j0 || e == j1) ? gs[e] : 0.f;
  if (b == 0) {
    float a = 0.f;
#pragma unroll
    for (int e = 0; e < E_; ++e) a += rawmean[e] * maskmean[e];
    auxp[0] = 0.01f * (a / (float)E_) * (float)(E_ * E_);
  }
}

// ---------------------------------------------------------------------------
// GEMM 1: xz = x @ in_w.T + in_b   (M=1568, N=1536, K=768) via WMMA f32.
// Register-blocked: one wave computes 32(M) x 64(N) = 2x4 WMMA tiles,
// so 6 fragment loads feed 8 WMMAs (0.75 loads/wmma).
//   xi (c<768) -> [B,D,L] layout for the conv;  z (c>=768) -> [B,L,D]
// ---------------------------------------------------------------------------
__global__ __launch_bounds__(128) void gemm_xz_kernel(
    const float* __restrict__ x, const float* __restrict__ in_w,
    const float* __restrict__ in_b, float* __restrict__ xi,
    float* __restrict__ z) {
  const int lane = threadIdx.x & 31;
  const int wave = threadIdx.x >> 5;
  const int hi = lane >> 4;
  const int lm = lane & 15;
  const int mBase = blockIdx.x * 32;              // 49 tiles
  const int nBase = blockIdx.y * 256 + wave * 64; // 6 blocks * 4 waves
  const float* aR0 = x + (size_t)(mBase + lm) * DIM + 2 * hi;
  const float* aR1 = aR0 + (size_t)16 * DIM;
  const float* bR[4];
#pragma unroll
  for (int j = 0; j < 4; ++j)
    bR[j] = in_w + (size_t)(nBase + j * 16 + lm) * DIM + 2 * hi;
  __builtin_prefetch(aR0, 0, 3);
  __builtin_prefetch(aR1, 0, 3);
  __builtin_prefetch(bR[0], 0, 3);
  v8f acc[2][4];
#pragma unroll
  for (int mi = 0; mi < 2; ++mi)
#pragma unroll
    for (int j = 0; j < 4; ++j) acc[mi][j] = (v8f){};
#pragma unroll 4
  for (int k = 0; k < DIM; k += 4) {
    v2f a0 = {aR0[k], aR0[k + 1]};
    v2f a1 = {aR1[k], aR1[k + 1]};
#pragma unroll
    for (int j = 0; j < 4; ++j) {
      v2f b = {bR[j][k], bR[j][k + 1]};
      acc[0][j] = wmma_f32x4(a0, b, acc[0][j]);
      acc[1][j] = wmma_f32x4(a1, b, acc[1][j]);
    }
  }
#pragma unroll
  for (int j = 0; j < 4; ++j) {
    const int n = nBase + j * 16 + lm;
    const float bias = in_b[n];
#pragma unroll
    for (int mi = 0; mi < 2; ++mi)
#pragma unroll
      for (int v = 0; v < 8; ++v) {
        int mr = mBase + mi * 16 + v + 8 * hi;
        int bb = mr / L_, ll = mr % L_;
        float val = acc[mi][j][v] + bias;
        if (n < DIM)
          xi[((size_t)(bb * DIM + n)) * L_ + ll] = val;
        else
          z[(size_t)mr * DIM + (n - DIM)] = val;
      }
  }
}

// ---------------------------------------------------------------------------
// Depthwise 3x3 SAME conv + bias + SiLU :  xi[B,D,7,7] -> xconv[B,D,L]
// ---------------------------------------------------------------------------
__global__ __launch_bounds__(256) void conv_silu_kernel(
    const float* __restrict__ xi, const float* __restrict__ cw,
    const float* __restrict__ cb, float* __restrict__ xconv) {
  int idx = blockIdx.x * blockDim.x + threadIdx.x;
  if (idx >= B_ * DIM * L_) return;
  int l = idx % L_;
  int t = idx / L_;
  int d = t % DIM;
  int b = t / DIM;
  int h = l / 7, w = l % 7;
  const float* src = xi + ((size_t)(b * DIM + d)) * L_;
  const float* wg = cw + (size_t)d * 9;
  float s = 0.f;
#pragma unroll
  for (int kh = -1; kh <= 1; ++kh)
#pragma unroll
    for (int kw = -1; kw <= 1; ++kw) {
      int hh = h + kh, ww = w + kw;
      if (hh >= 0 && hh < 7 && ww >= 0 && ww < 7)
        s += src[hh * 7 + ww] * wg[(kh + 1) * 3 + (kw + 1)];
    }
  s += cb[d];
  xconv[idx] = silu_f(s);
}

// ---------------------------------------------------------------------------
// GEMM 2: x_dbl[k,c,m] = sum_d xpw[k,c,d] * xs[k,d,m]  (M=80,N=1568,K=768)
// Wave computes 16 x 64 (1x4 tiles); xs permutation folded into addressing.
// ---------------------------------------------------------------------------
__global__ __launch_bounds__(128) void gemm_xdbl_kernel(
    const float* __restrict__ xpw, const float* __restrict__ xconv,
    float* __restrict__ x_dbl) {
  const int kdir = blockIdx.z;
  const int lane = threadIdx.x & 31;
  const int wave = threadIdx.x >> 5;
  const int hi = lane >> 4;
  const int lm = lane & 15;
  const int mBase = blockIdx.x * 16;              // 5 tiles
  const int nBase = blockIdx.y * 256 + wave * 64; // 7 blocks
  const float* aRow =
      xpw + ((size_t)kdir * CPROJ + mBase + lm) * DIM + 2 * hi;
  __builtin_prefetch(aRow, 0, 3);
  int nc[4];
  const float* bCol[4];
#pragma unroll
  for (int j = 0; j < 4; ++j) {
    nc[j] = nBase + j * 16 + lm;
    int n = nc[j] < BL ? nc[j] : (BL - 1);
    int bb = n / L_, ll = n % L_;
    bCol[j] = xconv + ((size_t)bb * DIM) * L_ + scan_idx(kdir, ll);
  }
  v8f acc[4];
#pragma unroll
  for (int j = 0; j < 4; ++j) acc[j] = (v8f){};
#pragma unroll 4
  for (int d0 = 0; d0 < DIM; d0 += 4) {
    const int d = d0 + 2 * hi;
    v2f a = {aRow[d0], aRow[d0 + 1]};
#pragma unroll
    for (int j = 0; j < 4; ++j) {
      v2f b = {bCol[j][(size_t)d * L_], bCol[j][(size_t)(d + 1) * L_]};
      acc[j] = wmma_f32x4(a, b, acc[j]);
    }
  }
#pragma unroll
  for (int j = 0; j < 4; ++j) {
    if (nc[j] < BL) {
#pragma unroll
      for (int v = 0; v < 8; ++v) {
        int mr = mBase + v + 8 * hi;
        x_dbl[((size_t)kdir * CPROJ + mr) * BL + nc[j]] = acc[j][v];
      }
    }
  }
}

// ---------------------------------------------------------------------------
// GEMM 3: dt[k,m,d] = softplus(sum_r dtw[k,d,r]*x_dbl[k,r,m] + dtb[k,d])
// (M=768, N=1568, K=48); wave computes 32 x 64; output [K,BL,D] for scan.
// ---------------------------------------------------------------------------
__global__ __launch_bounds__(128) void gemm_dt_kernel(
    const float* __restrict__ dtw, const float* __restrict__ dtb,
    const float* __restrict__ x_dbl, float* __restrict__ dtv) {
  const int kdir = blockIdx.z;
  const int lane = threadIdx.x & 31;
  const int wave = threadIdx.x >> 5;
  const int hi = lane >> 4;
  const int lm = lane & 15;
  const int mBase = blockIdx.x * 32;              // 24 tiles
  const int nBase = blockIdx.y * 256 + wave * 64; // 7 blocks
  const float* aR0 = dtw + ((size_t)kdir * DIM + mBase + lm) * R_ + 2 * hi;
  const float* aR1 = aR0 + (size_t)16 * R_;
  const float* bBase = x_dbl + (size_t)kdir * CPROJ * BL;
  int nc[4], nld[4];
#pragma unroll
  for (int j = 0; j < 4; ++j) {
    nc[j] = nBase + j * 16 + lm;
    nld[j] = nc[j] < BL ? nc[j] : (BL - 1);
  }
  v8f acc[2][4];
#pragma unroll
  for (int mi = 0; mi < 2; ++mi)
#pragma unroll
    for (int j = 0; j < 4; ++j) acc[mi][j] = (v8f){};
#pragma unroll
  for (int r0 = 0; r0 < R_; r0 += 4) {
    const int r = r0 + 2 * hi;
    v2f a0 = {aR0[r0], aR0[r0 + 1]};
    v2f a1 = {aR1[r0], aR1[r0 + 1]};
#pragma unroll
    for (int j = 0; j < 4; ++j) {
      v2f b = {bBase[(size_t)r * BL + nld[j]],
               bBase[(size_t)(r + 1) * BL + nld[j]]};
      acc[0][j] = wmma_f32x4(a0, b, acc[0][j]);
      acc[1][j] = wmma_f32x4(a1, b, acc[1][j]);
    }
  }
#pragma unroll
  for (int j = 0; j < 4; ++j) {
    if (nc[j] < BL) {
#pragma unroll
      for (int mi = 0; mi < 2; ++mi)
#pragma unroll
        for (int v = 0; v < 8; ++v) {
          int mr = mBase + mi * 16 + v + 8 * hi;
          float t = acc[mi][j][v] + dtb[kdir * DIM + mr];
          float sp = (t > 20.f) ? t : log1pf(__expf(t));
          dtv[((size_t)kdir * BL + nc[j]) * DIM + mr] = sp;
        }
    }
  }
}

// ---------------------------------------------------------------------------
// Selective scan: one thread per channel d; 16-state recurrence in VGPRs,
// B/C sequences staged in LDS.  ys[K,B,L,D] = y + Ds*u
// ---------------------------------------------------------------------------
__global__ __launch_bounds__(768) void scan_kernel(
    const float* __restrict__ x_dbl, const float* __restrict__ dtv,
    const float* __restrict__ A_log, const float* __restrict__ Ds,
    const float* __restrict__ xconv, float* __restrict__ ys) {
  const int b = blockIdx.x;
  const int kdir = blockIdx.y;
  const int d = threadIdx.x;
  __shared__ float Bl[N_][L_ + 1];
  __shared__ float Cl[N_][L_ + 1];
  for (int i = threadIdx.x; i < N_ * L_; i += blockDim.x) {
    int n = i / L_, l = i % L_;
    Bl[n][l] = x_dbl[((size_t)kdir * CPROJ + R_ + n) * BL + b * L_ + l];
    Cl[n][l] = x_dbl[((size_t)kdir * CPROJ + R_ + N_ + n) * BL + b * L_ + l];
  }
  __syncthreads();
  float Areg[N_], h[N_];
#pragma unroll
  for (int n = 0; n < N_; ++n) {
    Areg[n] = -__expf(A_log[((size_t)kdir * DIM + d) * N_ + n]);
    h[n] = 0.f;
  }
  const float Dsd = Ds[kdir * DIM + d];
  const float* uBase = xconv + (size_t)(b * DIM + d) * L_;
  const float* dtBase = dtv + ((size_t)kdir * BL + b * L_) * DIM + d;
  float* yOut = ys + ((size_t)(kdir * B_ + b) * L_) * DIM + d;
  for (int l = 0; l < L_; ++l) {
    float dt = dtBase[(size_t)l * DIM];
    float u = uBase[scan_idx(kdir, l)];
    float du = dt * u;
    float y = 0.f;
#pragma unroll
    for (int n = 0; n < N_; ++n) {
      h[n] = __expf(dt * Areg[n]) * h[n] + du * Bl[n][l];
      y += h[n] * Cl[n][l];
    }
    yOut[(size_t)l * DIM] = y + Dsd * u;
  }
}

// ---------------------------------------------------------------------------
// Combine 4 directions -> LayerNorm(768) -> * SiLU(z) :  yg[B,L,D]
// ---------------------------------------------------------------------------
__global__ __launch_bounds__(256) void combine_ln_kernel(
    const float* __restrict__ ys, const float* __restrict__ z,
    const float* __restrict__ onw, const float* __restrict__ onb,
    float* __restrict__ yg) {
  const int bl = blockIdx.x;
  const int b = bl / L_, l = bl % L_;
  const int t = threadIdx.x;
  __shared__ float yt[DIM];
  __shared__ float red[256], red2[256];
  const int pv = (l % 7) * 7 + l / 7;
  const float* y0 = ys + ((size_t)(0 * B_ + b) * L_ + l) * DIM;
  const float* y1 = ys + ((size_t)(1 * B_ + b) * L_ + (48 - l)) * DIM;
  const float* y2 = ys + ((size_t)(2 * B_ + b) * L_ + pv) * DIM;
  const float* y3 = ys + ((size_t)(3 * B_ + b) * L_ + (48 - pv)) * DIM;
  float s = 0.f, s2 = 0.f;
  for (int d = t; d < DIM; d += 256) {
    float v = y0[d] + y1[d] + y2[d] + y3[d];
    yt[d] = v;
    s += v;
    s2 += v * v;
  }
  red[t] = s;
  red2[t] = s2;
  __syncthreads();
  for (int off = 128; off > 0; off >>= 1) {
    if (t < off) { red[t] += red[t + off]; red2[t] += red2[t + off]; }
    __syncthreads();
  }
  const float mean = red[0] / (float)DIM;
  const float var = red2[0] / (float)DIM - mean * mean;
  const float inv = rsqrtf(var + 1e-5f);
  const float* zrow = z + (size_t)(b * L_ + l) * DIM;
  float* orow = yg + (size_t)(b * L_ + l) * DIM;
  for (int d = t; d < DIM; d += 256) {
    float yn = (yt[d] - mean) * inv * onw[d] + onb[d];
    orow[d] = yn * silu_f(zrow[d]);
  }
}

// Average pool over L (deterministic, no fp atomics)
__global__ __launch_bounds__(256) void pool_kernel(const float* __restrict__ yg,
                                                   float* __restrict__ feat) {
  int idx = blockIdx.x * blockDim.x + threadIdx.x;
  if (idx >= B_ * DIM) return;
  int b = idx / DIM, d = idx % DIM;
  float s = 0.f;
  for (int l = 0; l < L_; ++l) s += yg[(size_t)(b * L_ + l) * DIM + d];
  feat[idx] = s / (float)L_;
}

// Final per-sample LayerNorm of pooled features -> expert_out[e][B][D]
__global__ __launch_bounds__(256) void ln_feat_kernel(
    const float* __restrict__ feat, const float* __restrict__ lw,
    const float* __restrict__ lb, float* __restrict__ eo) {
  const int b = blockIdx.x;
  const int t = threadIdx.x;
  __shared__ float red[256], red2[256];
  float s = 0.f, s2 = 0.f;
  for (int d = t; d < DIM; d += 256) {
    float v = feat[b * DIM + d];
    s += v;
    s2 += v * v;
  }
  red[t] = s;
  red2[t] = s2;
  __syncthreads();
  for (int off = 128; off > 0; off >>= 1) {
    if (t < off) { red[t] += red[t + off]; red2[t] += red2[t + off]; }
    __syncthreads();
  }
  const float mean = red[0] / (float)DIM;
  const float var = red2[0] / (float)DIM - mean * mean;
  const float inv = rsqrtf(var + 1e-5f);
  for (int d = t; d < DIM; d += 256) {
    float v = feat[b * DIM + d];
    eo[(size_t)b * DIM + d] = (v - mean) * inv * lw[d] + lb[d];
  }
}

// mixed[b,d] = sum_e w[b,e] * expert_out[e,b,d];  aux appended at out[B*DIM]
__global__ __launch_bounds__(256) void mix_kernel(
    const float* __restrict__ w, const float* __restrict__ eo,
    const float* __restrict__ auxp, float* __restrict__ out) {
  int idx = blockIdx.x * blockDim.x + threadIdx.x;
  if (idx < B_ * DIM) {
    int b = idx / DIM, d = idx % DIM;
    float s = 0.f;
#pragma unroll
    for (int e = 0; e < E_; ++e)
      s += w[b * E_ + e] * eo[((size_t)e * B_ + b) * DIM + d];
    out[idx] = s;
  }
  if (idx == 0) out[B_ * DIM] = auxp[0];
}

// ---------------------------------------------------------------------------
extern "C" void kernel_launch(void* const* d_in, const int* in_sizes, int n_in,
                              void* d_out, int out_size, void* d_ws,
                              size_t ws_size, hipStream_t stream) {
  const float* x         = (const float*)d_in[0];
  const float* gate_w    = (const float*)d_in[1];
  const float* gate_b    = (const float*)d_in[2];
  const float* in_proj_w = (const float*)d_in[3];
  const float* in_proj_b = (const float*)d_in[4];
  const float* conv_w    = (const float*)d_in[5];
  const float* conv_b    = (const float*)d_in[6];
  const float* x_proj_w  = (const float*)d_in[7];
  const float* dt_proj_w = (const float*)d_in[8];
  const float* dt_proj_b = (const float*)d_in[9];
  const float* A_log     = (const float*)d_in[10];
  const float* Ds        = (const float*)d_in[11];
  const float* out_nw    = (const float*)d_in[12];
  const float* out_nb    = (const float*)d_in[13];
  const float* ln_w      = (const float*)d_in[14];
  const float* ln_b      = (const float*)d_in[15];
  float* out = (float*)d_out;

  float* ws = (float*)d_ws;
  size_t o = 0;
  float* xi    = ws + o; o += (size_t)B_ * DIM * L_;      // [B,D,L]
  float* xconv = ws + o; o += (size_t)B_ * DIM * L_;      // [B,D,L]
  float* zbuf  = ws + o; o += (size_t)B_ * L_ * DIM;      // [B,L,D]
  float* xdbl  = ws + o; o += (size_t)K_ * CPROJ * BL;    // [K,80,BL]
  float* dtv   = ws + o; o += (size_t)K_ * BL * DIM;      // [K,BL,D]
  float* ysb   = ws + o; o += (size_t)K_ * B_ * L_ * DIM; // [K,B,L,D]
  float* yg    = ws + o; o += (size_t)B_ * L_ * DIM;      // [B,L,D]
  float* feat  = ws + o; o += (size_t)B_ * DIM;
  float* eout  = ws + o; o += (size_t)E_ * B_ * DIM;
  float* raw   = ws + o; o += B_ * E_;
  float* wbuf  = ws + o; o += B_ * E_;
  float* auxb  = ws + o; o += 1;
  (void)ws_size; (void)in_sizes; (void)n_in; (void)out_size;

  gate1_kernel<<<B_, 256, 0, stream>>>(x, gate_w, gate_b, raw);
  gate2_kernel<<<1, 32, 0, stream>>>(raw, wbuf, auxb);

  for (int e = 0; e < E_; ++e) {
    const float* in_w_e = in_proj_w + (size_t)e * D2 * DIM;
    const float* in_b_e = in_proj_b + (size_t)e * D2;
    const float* cw_e   = conv_w + (size_t)e * DIM * 9;
    const float* cb_e   = conv_b + (size_t)e * DIM;
    const float* xpw_e  = x_proj_w + (size_t)e * K_ * CPROJ * DIM;
    const float* dtw_e  = dt_proj_w + (size_t)e * K_ * DIM * R_;
    const float* dtb_e  = dt_proj_b + (size_t)e * K_ * DIM;
    const float* Al_e   = A_log + (size_t)e * K_ * DIM * N_;
    const float* Ds_e   = Ds + (size_t)e * K_ * DIM;
    const float* onw_e  = out_nw + (size_t)e * DIM;
    const float* onb_e  = out_nb + (size_t)e * DIM;
    const float* lnw_e  = ln_w + (size_t)e * DIM;
    const float* lnb_e  = ln_b + (size_t)e * DIM;
    float* eo_e = eout + (size_t)e * B_ * DIM;

    gemm_xz_kernel<<<dim3(BL / 32, D2 / 256), 128, 0, stream>>>(
        x, in_w_e, in_b_e, xi, zbuf);
    conv_silu_kernel<<<(B_ * DIM * L_ + 255) / 256, 256, 0, stream>>>(
        xi, cw_e, cb_e, xconv);
    gemm_xdbl_kernel<<<dim3(CPROJ / 16, (BL + 255) / 256, K_), 128, 0,
                       stream>>>(xpw_e, xconv, xdbl);
    gemm_dt_kernel<<<dim3(DIM / 32, (BL + 255) / 256, K_), 128, 0, stream>>>(
        dtw_e, dtb_e, xdbl, dtv);
    scan_kernel<<<dim3(B_, K_), 768, 0, stream>>>(xdbl, dtv, Al_e, Ds_e, xconv,
                                                  ysb);
    combine_ln_kernel<<<B_ * L_, 256, 0, stream>>>(ysb, zbuf, onw_e, onb_e, yg);
    pool_kernel<<<(B_ * DIM + 255) / 256, 256, 0, stream>>>(yg, feat);
    ln_feat_kernel<<<B_, 256, 0, stream>>>(feat, lnw_e, lnb_e, eo_e);
  }
  mix_kernel<<<(B_ * DIM + 255) / 256, 256, 0, stream>>>(wbuf, eout, auxb, out);
}